// UnitaryDevelopmentLayer_13451837571104
// MI455X (gfx1250) — compile-verified
//
#include <hip/hip_runtime.h>

typedef __attribute__((ext_vector_type(2))) float v2f;
typedef __attribute__((ext_vector_type(8))) float v8f;

#define NWAVES 8
#define TSTEPS 256   // 2048 / 8 waves
#define SEQ_L  2049
#define DD     8
#define CC     2
#define NSQ    3     // squarings; dx pre-scaled by 2^-3
#define KORD   9     // Taylor order

// One 16x16x4 f32 MAC step: D = A(16x4) x B(4x16) + C
__device__ __forceinline__ v8f wmma4(v2f a, v2f b, v8f c) {
  return __builtin_amdgcn_wmma_f32_16x16x4_f32(
      /*neg_a=*/false, a, /*neg_b=*/false, b,
      /*c_mod=*/(short)0, c, /*reuse_a=*/false, /*reuse_b=*/false);
}

// Cross-half lane swap (lane ^ 16) via ds_swizzle SWAPX16 (no LDS memory).
__device__ __forceinline__ float swap16(float v) {
  return __int_as_float(__builtin_amdgcn_ds_swizzle(__float_as_int(v), 0x401F));
}

// C/D layout (row = v + 8*hi, col = lo)  ->  B-operand chunks
// (row = 4j + u + 2*hi, col = lo).  4 swizzles + 16 cndmask per matrix.
__device__ __forceinline__ void cd_to_B(const v8f& M, v2f B[4], int hi) {
  const bool h = (hi != 0);
#pragma unroll
  for (int u = 0; u < 2; ++u) {
    {  // regs {u, 2+u} -> B[0][u], B[2][u]
      const float y = h ? M[2 + u] : M[u];
      const float z = h ? M[u] : M[2 + u];
      const float sz = swap16(z);
      B[0][u] = h ? sz : y;
      B[2][u] = h ? y : sz;
    }
    {  // regs {4+u, 6+u} -> B[1][u], B[3][u]
      const float y = h ? M[6 + u] : M[4 + u];
      const float z = h ? M[4 + u] : M[6 + u];
      const float sz = swap16(z);
      B[1][u] = h ? sz : y;
      B[3][u] = h ? y : sz;
    }
  }
}

// Store C/D accumulator to row-major LDS (true transpose path).
__device__ __forceinline__ void stage_cd(float* buf, const v8f& M, int lo, int hi) {
#pragma unroll
  for (int v = 0; v < 8; ++v) buf[(v + hi * 8) * 16 + lo] = M[v];
}

// Row-major LDS -> A-operand chunks (row=lo, K=4j+u+2hi): one b64 per chunk.
__device__ __forceinline__ void load_Aop(const float* buf, v2f A[4], int lo, int hi) {
#pragma unroll
  for (int j = 0; j < 4; ++j)
    A[j] = *(const v2f*)(buf + lo * 16 + 4 * j + 2 * hi);
}

// Row-major LDS -> B-operand chunks (combine phase only).
__device__ __forceinline__ void load_Bop(const float* buf, v2f B[4], int lo, int hi) {
#pragma unroll
  for (int j = 0; j < 4; ++j) {
    B[j][0] = buf[(4 * j + 2 * hi + 0) * 16 + lo];
    B[j][1] = buf[(4 * j + 2 * hi + 1) * 16 + lo];
  }
}

// Row-major LDS -> C/D layout registers (combine phase only).
__device__ __forceinline__ void load_cd(const float* buf, v8f& M, int lo, int hi) {
#pragma unroll
  for (int v = 0; v < 8; ++v) M[v] = buf[(v + hi * 8) * 16 + lo];
}

__device__ __forceinline__ void sum4(const v2f a[4], const v2f b[4], v2f s[4]) {
#pragma unroll
  for (int j = 0; j < 4; ++j) s[j] = a[j] + b[j];
}

// 3M complex 16x16 matmul: 12 WMMAs.
// T3=(Ar+Ai)(Br+Bi), T1=ArBr, T2=AiBi; Dr=T1-T2, Di=T3-T1-T2.
__device__ __forceinline__ void cmm3(const v2f Ar[4], const v2f Ai[4], const v2f As[4],
                                     const v2f Br[4], const v2f Bi[4], const v2f Bs[4],
                                     v8f& Dr, v8f& Di) {
  const v8f z = {0.f, 0.f, 0.f, 0.f, 0.f, 0.f, 0.f, 0.f};
  v8f t3 = z, t1 = z, t2 = z;
#pragma unroll
  for (int j = 0; j < 4; ++j) t3 = wmma4(As[j], Bs[j], t3);
#pragma unroll
  for (int j = 0; j < 4; ++j) t1 = wmma4(Ar[j], Br[j], t1);
  v8f di;
#pragma unroll
  for (int v = 0; v < 8; ++v) di[v] = t3[v] - t1[v];
#pragma unroll
  for (int j = 0; j < 4; ++j) t2 = wmma4(Ai[j], Bi[j], t2);
#pragma unroll
  for (int v = 0; v < 8; ++v) {
    Dr[v] = t1[v] - t2[v];
    Di[v] = di[v] - t2[v];
  }
}

__global__ void __launch_bounds__(256, 1)
__attribute__((amdgpu_num_vgpr(256)))
udl_kernel(const float* __restrict__ x,
           const float* __restrict__ A_re,
           const float* __restrict__ A_im,
           float* __restrict__ out) {
  // Packed transposed generators: quad (re_q, re_q1, im_q, im_q1) for
  // T[r][q..q+1], T = Aah^T, so the A-layout G build is one b128 per chunk.
  __shared__ __align__(16) float aahQ[CC][DD][128][4];
  __shared__ __align__(16) float stg[NWAVES][2][256];  // per-wave staging
  __shared__ __align__(16) float comb[2][256];         // wave-0 combine staging

  const int tid = threadIdx.x;

  for (int idx = tid; idx < CC * DD * 256; idx += 256) {
    const int e = idx & 255;
    const int d = (idx >> 8) & 7;
    const int c = idx >> 11;
    const int i = e >> 4, j = e & 15;   // Aah element (row=i, col=j)
    const float* re = A_re + (size_t)(d * CC + c) * 256;
    const float* im = A_im + (size_t)(d * CC + c) * 256;
    const float vr = 0.5f * (re[i * 16 + j] - re[j * 16 + i]);
    const float vi = 0.5f * (im[i * 16 + j] + im[j * 16 + i]);
    // transposed: T[r=j][q=i]
    aahQ[c][d][j * 8 + (i >> 1)][(i & 1)]     = vr;
    aahQ[c][d][j * 8 + (i >> 1)][2 + (i & 1)] = vi;
  }
  __syncthreads();

  const int wave = tid >> 5;
  const int lane = tid & 31;
  const int lo = lane & 15;
  const int hi = lane >> 4;
  const int n = blockIdx.x >> 1;
  const int c = blockIdx.x & 1;

  float ident[8];
#pragma unroll
  for (int v = 0; v < 8; ++v)
    ident[v] = (lane == v || lane == v + 24) ? 1.0f : 0.0f;

  float* stre = &stg[wave][0][0];
  float* stim = &stg[wave][1][0];

  // running transposed chunk product Q = P^T, updated Q <- M_t^T @ Q
  v8f Qr, Qi;
#pragma unroll
  for (int v = 0; v < 8; ++v) { Qr[v] = ident[v]; Qi[v] = 0.0f; }

  const int t0 = wave * TSTEPS;
  const float* xbase = x + (size_t)n * SEQ_L * DD;

  float xp[DD];
#pragma unroll
  for (int d = 0; d < DD; ++d) xp[d] = xbase[(size_t)t0 * DD + d];

  for (int t = t0; t < t0 + TSTEPS; ++t) {
    // fence the scheduler: do not pipeline the next step's generator build
    // into the current expm chain (it only inflates VGPR liveness).
    __builtin_amdgcn_sched_barrier(0);

    float dxs[DD];
#pragma unroll
    for (int d = 0; d < DD; ++d) {
      const float xn = xbase[(size_t)(t + 1) * DD + d];
      dxs[d] = (xn - xp[d]) * 0.125f;  // 2^-NSQ scaling for expm
      xp[d] = xn;
    }

    // X = G^T / 8 built directly in A-operand layout (b128 LDS loads)
    v2f Gr[4], Gi[4];
#pragma unroll
    for (int j = 0; j < 4; ++j) {
      Gr[j][0] = 0.f; Gr[j][1] = 0.f;
      Gi[j][0] = 0.f; Gi[j][1] = 0.f;
    }
#pragma unroll
    for (int d = 0; d < DD; ++d) {
      const float4* tp = (const float4*)&aahQ[c][d][0][0];
#pragma unroll
      for (int j = 0; j < 4; ++j) {
        const float4 q = tp[lo * 8 + 2 * j + hi];
        Gr[j][0] += dxs[d] * q.x; Gr[j][1] += dxs[d] * q.y;
        Gi[j][0] += dxs[d] * q.z; Gi[j][1] += dxs[d] * q.w;
      }
    }
    v2f Gs[4];
    sum4(Gr, Gi, Gs);  // hoisted 3M A-side sum for the whole Horner loop

    // expm(X): coefficient-form Horner, R = X*R + a_k*I  (a_k = 1/k!)
    v8f Rr, Ri;
#pragma unroll
    for (int v = 0; v < 8; ++v) {
      Rr[v] = ident[v] * (1.0f / 362880.0f);  // a_9
      Ri[v] = 0.0f;
    }
#pragma unroll
    for (int k = KORD - 1; k >= 0; --k) {
      v2f Br[4], Bi[4], Bs[4];
      cd_to_B(Rr, Br, hi);
      cd_to_B(Ri, Bi, hi);
      sum4(Br, Bi, Bs);
      v8f Tr, Ti;
      cmm3(Gr, Gi, Gs, Br, Bi, Bs, Tr, Ti);
      const float a =
          (k == 0) ? 1.f : (k == 1) ? 1.f : (k == 2) ? 0.5f :
          (k == 3) ? (1.f / 6.f) : (k == 4) ? (1.f / 24.f) :
          (k == 5) ? (1.f / 120.f) : (k == 6) ? (1.f / 720.f) :
          (k == 7) ? (1.f / 5040.f) : (1.f / 40320.f);
#pragma unroll
      for (int v = 0; v < 8; ++v) Rr[v] = Tr[v] + a * ident[v];
      Ri = Ti;  // imaginary part: pure rename
    }

    // NSQ squarings: rolled loop (serially dependent) to cap VGPR pressure
#pragma unroll 1
    for (int sq = 0; sq < NSQ; ++sq) {
      stage_cd(stre, Rr, lo, hi);
      stage_cd(stim, Ri, lo, hi);
      v2f Br[4], Bi[4], Bs[4];
      cd_to_B(Rr, Br, hi);
      cd_to_B(Ri, Bi, hi);
      sum4(Br, Bi, Bs);
      v2f Ar[4], Ai[4], As[4];
      load_Aop(stre, Ar, lo, hi);
      load_Aop(stim, Ai, lo, hi);
      sum4(Ar, Ai, As);
      cmm3(Ar, Ai, As, Br, Bi, Bs, Rr, Ri);
    }

    // Q <- M_t^T @ Q  (R = M_t^T as left operand, Q right via swizzle)
    {
      stage_cd(stre, Rr, lo, hi);
      stage_cd(stim, Ri, lo, hi);
      v2f Br[4], Bi[4], Bs[4];
      cd_to_B(Qr, Br, hi);
      cd_to_B(Qi, Bi, hi);
      sum4(Br, Bi, Bs);
      v2f Ar[4], Ai[4], As[4];
      load_Aop(stre, Ar, lo, hi);
      load_Aop(stim, Ai, lo, hi);
      sum4(Ar, Ai, As);
      cmm3(Ar, Ai, As, Br, Bi, Bs, Qr, Qi);
    }
  }

  // publish each wave's transposed chunk product (row-major)
  stage_cd(stre, Qr, lo, hi);
  stage_cd(stim, Qi, lo, hi);
  __syncthreads();

  if (wave == 0) {
    // Q_total = Q7 @ Q6 @ ... @ Q0  (equals (P0 P1 ... P7)^T)
    v8f accr, acci;
    load_cd(&stg[7][0][0], accr, lo, hi);
    load_cd(&stg[7][1][0], acci, lo, hi);
    for (int w = 6; w >= 1; --w) {
      stage_cd(&comb[0][0], accr, lo, hi);
      stage_cd(&comb[1][0], acci, lo, hi);
      v2f Ar[4], Ai[4], As[4], Br[4], Bi[4], Bs[4];
      load_Aop(&comb[0][0], Ar, lo, hi);
      load_Aop(&comb[1][0], Ai, lo, hi);
      sum4(Ar, Ai, As);
      load_Bop(&stg[w][0][0], Br, lo, hi);
      load_Bop(&stg[w][1][0], Bi, lo, hi);
      sum4(Br, Bi, Bs);
      cmm3(Ar, Ai, As, Br, Bi, Bs, accr, acci);
    }
    {  // final right factor: wave 0's own Q (still in registers)
      stage_cd(&comb[0][0], accr, lo, hi);
      stage_cd(&comb[1][0], acci, lo, hi);
      v2f Ar[4], Ai[4], As[4], Br[4], Bi[4], Bs[4];
      load_Aop(&comb[0][0], Ar, lo, hi);
      load_Aop(&comb[1][0], Ai, lo, hi);
      sum4(Ar, Ai, As);
      cd_to_B(Qr, Br, hi);
      cd_to_B(Qi, Bi, hi);
      sum4(Br, Bi, Bs);
      cmm3(Ar, Ai, As, Br, Bi, Bs, accr, acci);
    }
    // acc = P^T: lane holds acc[v] = P[lo][v+8hi] -> transpose on store.
    float* op = out + (size_t)(n * CC + c) * 512;
#pragma unroll
    for (int v = 0; v < 8; ++v) {
      const int col = v + hi * 8;
      op[(lo * 16 + col) * 2 + 0] = accr[v];
      op[(lo * 16 + col) * 2 + 1] = acci[v];
    }
  }
}

extern "C" void kernel_launch(void* const* d_in, const int* in_sizes, int n_in,
                              void* d_out, int out_size, void* d_ws, size_t ws_size,
                              hipStream_t stream) {
  const float* x    = (const float*)d_in[0];   // (64, 2049, 8)
  const float* A_re = (const float*)d_in[1];   // (8, 2, 16, 16)
  const float* A_im = (const float*)d_in[2];   // (8, 2, 16, 16)
  float* out = (float*)d_out;                  // (64, 2, 16, 16) complex64 interleaved
  (void)in_sizes; (void)n_in; (void)out_size; (void)d_ws; (void)ws_size;
  udl_kernel<<<dim3(64 * CC), dim3(256), 0, stream>>>(x, A_re, A_im, out);
}